// GNN_15109694948150
// MI455X (gfx1250) — compile-verified
//
#include <hip/hip_runtime.h>

typedef __attribute__((ext_vector_type(2))) float v2f;
typedef __attribute__((ext_vector_type(8))) float v8f;

#define NNODES 50000
#define NEDGES 800000

// ---------------- degree / dinv ----------------

__global__ void init_deg_k(float* deg, int n) {
    int i = blockIdx.x * blockDim.x + threadIdx.x;
    if (i < n) deg[i] = 1.0f;  // self-loop
}

__global__ void edge_deg_k(const long long* __restrict__ col, float* __restrict__ deg, int ne) {
    int i = blockIdx.x * blockDim.x + threadIdx.x;
    if (i < ne) unsafeAtomicAdd(&deg[(int)col[i]], 1.0f);
}

__global__ void make_dinv_k(float* deg, int n) {
    int i = blockIdx.x * blockDim.x + threadIdx.x;
    if (i < n) deg[i] = rsqrtf(deg[i]);   // deg >= 1 always (self loops)
}

// ---------------- fp32 WMMA GEMM: C[M,N] = A[M,K] @ B[K,N] ----------------
// One wave computes one 16x16 tile of C via V_WMMA_F32_16X16X4_F32.
// blockDim = (32, N/16): wave y handles N-tile y. M must be a multiple of 16.

__global__ void gemm_wmma_f32(const float* __restrict__ A,
                              const float* __restrict__ B,
                              float* __restrict__ C,
                              int K, int N) {
    const int l    = threadIdx.x & 15;   // 0..15
    const int half = threadIdx.x >> 4;   // 0 or 1
    const int m0   = blockIdx.x * 16;
    const int n0   = threadIdx.y * 16;

    v8f acc = {};
    const float* arow = A + (size_t)(m0 + l) * K + 2 * half;  // lanes 16-31: K offset +2
    for (int k0 = 0; k0 < K; k0 += 4) {
        v2f a, b;
        // A 16x4: VGPR0 = K=k0(+2*half), VGPR1 = K=k0+1(+2*half)
        a.x = arow[k0];
        a.y = arow[k0 + 1];
        // B 4x16: VGPRv lanes0-15 -> K=k0+v, lanes16-31 -> K=k0+v+2
        const float* bp = B + (size_t)(k0 + 2 * half) * N + n0 + l;
        b.x = bp[0];
        b.y = bp[N];
        acc = __builtin_amdgcn_wmma_f32_16x16x4_f32(
            /*neg_a=*/false, a, /*neg_b=*/false, b,
            /*c_mod=*/(short)0, acc, /*reuse_a=*/false, /*reuse_b=*/false);
    }
    // C/D layout: VGPR v -> row (v + 8*half), col l
    for (int v = 0; v < 8; ++v)
        C[(size_t)(m0 + v + 8 * half) * N + n0 + l] = acc[v];
}

// ---------------- edge scatter: agg[col] += dinv[row]*dinv[col] * H[row] ----------------
// One thread per (edge, 4-float group). lg = log2(F/4).

__global__ void scatter_edges_k(const float* __restrict__ H,
                                float* __restrict__ agg,
                                const float* __restrict__ dinv,
                                const long long* __restrict__ row,
                                const long long* __restrict__ col,
                                int ne, int F, int lg) {
    long long gid = (long long)blockIdx.x * blockDim.x + threadIdx.x;
    long long e = gid >> lg;
    if (e >= ne) return;
    int g = (int)(gid & ((1 << lg) - 1));
    int r = (int)row[e];
    int c = (int)col[e];
    float nrm = dinv[r] * dinv[c];
    float4 h = *reinterpret_cast<const float4*>(H + (size_t)r * F + g * 4);
    float* ap = agg + (size_t)c * F + g * 4;
    unsafeAtomicAdd(ap + 0, nrm * h.x);
    unsafeAtomicAdd(ap + 1, nrm * h.y);
    unsafeAtomicAdd(ap + 2, nrm * h.z);
    unsafeAtomicAdd(ap + 3, nrm * h.w);
}

// ---------------- per-node epilogue: self-loop + bias (+ReLU), in place ----------------

__global__ void epilogue_k(const float* __restrict__ H,
                           float* __restrict__ agg_io,
                           const float* __restrict__ bias,
                           const float* __restrict__ dinv,
                           long long total, int lgF, int do_relu) {
    long long i = (long long)blockIdx.x * blockDim.x + threadIdx.x;
    if (i >= total) return;
    int node = (int)(i >> lgF);
    int f    = (int)(i & ((1 << lgF) - 1));
    float s = dinv[node];
    float v = agg_io[i] + s * s * H[i] + bias[f];
    if (do_relu) v = fmaxf(v, 0.0f);
    agg_io[i] = v;
}

// ---------------- launch ----------------

extern "C" void kernel_launch(void* const* d_in, const int* in_sizes, int n_in,
                              void* d_out, int out_size, void* d_ws, size_t ws_size,
                              hipStream_t stream) {
    const float*     x   = (const float*)d_in[0];
    const long long* ei  = (const long long*)d_in[1];   // int64 [2, E]
    const float*     W1  = (const float*)d_in[2];
    const float*     b1  = (const float*)d_in[3];
    const float*     W2  = (const float*)d_in[4];
    const float*     b2  = (const float*)d_in[5];
    float*           out = (float*)d_out;

    const long long* row = ei;
    const long long* col = ei + NEDGES;

    // workspace layout (floats)
    float* ws   = (float*)d_ws;
    float* dinv = ws;                       // 50000 (padded to 51200)
    float* H1   = ws + 51200;               // 50000*128 = 6,400,000
    float* agg1 = H1 + 6400000;             // 6,400,000 (relu'd in place -> layer-2 input)
    float* H2   = agg1 + 6400000;           // 50000*64 = 3,200,000

    // 1) degrees -> dinv
    init_deg_k<<<(NNODES + 255) / 256, 256, 0, stream>>>(dinv, NNODES);
    edge_deg_k<<<(NEDGES + 255) / 256, 256, 0, stream>>>(col, dinv, NEDGES);
    make_dinv_k<<<(NNODES + 255) / 256, 256, 0, stream>>>(dinv, NNODES);

    // 2) layer 1: H1 = X @ W1   (K=100, N=128) — fp32 WMMA
    gemm_wmma_f32<<<dim3(NNODES / 16), dim3(32, 8), 0, stream>>>(x, W1, H1, 100, 128);

    hipMemsetAsync(agg1, 0, (size_t)NNODES * 128 * sizeof(float), stream);
    scatter_edges_k<<<(int)(((long long)NEDGES * 32 + 255) / 256), 256, 0, stream>>>(
        H1, agg1, dinv, row, col, NEDGES, 128, 5);
    epilogue_k<<<(int)(((long long)NNODES * 128 + 255) / 256), 256, 0, stream>>>(
        H1, agg1, b1, dinv, (long long)NNODES * 128, 7, 1);

    // 3) layer 2: H2 = relu(agg1) @ W2   (K=128, N=64) — fp32 WMMA
    gemm_wmma_f32<<<dim3(NNODES / 16), dim3(32, 4), 0, stream>>>(agg1, W2, H2, 128, 64);

    hipMemsetAsync(out, 0, (size_t)NNODES * 64 * sizeof(float), stream);
    scatter_edges_k<<<(int)(((long long)NEDGES * 16 + 255) / 256), 256, 0, stream>>>(
        H2, out, dinv, row, col, NEDGES, 64, 4);
    epilogue_k<<<(int)(((long long)NNODES * 64 + 255) / 256), 256, 0, stream>>>(
        H2, out, b2, dinv, (long long)NNODES * 64, 6, 0);
}